// T2SDecodeStepWrapper_25735444037689
// MI455X (gfx1250) — compile-verified
//
#include <hip/hip_runtime.h>

#define N_LAYERS 24
#define HID      512
#define HEADS    16
#define HDIM     32
#define CAP      2048
#define VOCAB    1025
#define FF       2048

typedef __attribute__((ext_vector_type(2))) float v2f;
typedef __attribute__((ext_vector_type(4))) float v4f;
typedef __attribute__((ext_vector_type(8))) float v8f;

static const size_t KV_ELEMS = (size_t)N_LAYERS * CAP * HID;   // 25165824

// ---------------- wave / block reductions (wave32) ----------------
__device__ __forceinline__ float waveSum(float v) {
#pragma unroll
  for (int o = 16; o > 0; o >>= 1) v += __shfl_xor(v, o, 32);
  return v;
}
// reduce within each 16-lane half (enough when values are duplicated across halves)
__device__ __forceinline__ float half16Sum(float v) {
#pragma unroll
  for (int o = 8; o > 0; o >>= 1) v += __shfl_xor(v, o, 32);
  return v;
}
__device__ __forceinline__ float half16Max(float v) {
#pragma unroll
  for (int o = 8; o > 0; o >>= 1) v = fmaxf(v, __shfl_xor(v, o, 32));
  return v;
}
__device__ __forceinline__ float blockSum256(float v) {
  __shared__ float tmp[8];
  __shared__ float resv;
  v = waveSum(v);
  int w = threadIdx.x >> 5, l = threadIdx.x & 31;
  if (l == 0) tmp[w] = v;
  __syncthreads();
  if (threadIdx.x == 0) {
    float r = 0.f;
#pragma unroll
    for (int i = 0; i < 8; i++) r += tmp[i];
    resv = r;
  }
  __syncthreads();
  return resv;
}

// ---------------- bulk KV copy: keep pos<clen, zero pos>=clen (row clen rewritten later) ----
__global__ void kv_copy_kernel(const v4f* __restrict__ kin, const v4f* __restrict__ vin,
                               v4f* __restrict__ kout, v4f* __restrict__ vout,
                               const int* __restrict__ clen_p) {
  size_t idx = (size_t)blockIdx.x * 256 + threadIdx.x;   // v4f index, per tensor
  int pos = (int)((idx >> 7) & (CAP - 1));               // 128 v4f per row of 512 floats
  int clen = clen_p[0];
  v4f z = {0.f, 0.f, 0.f, 0.f};
  v4f kv = (pos < clen) ? __builtin_nontemporal_load(&kin[idx]) : z;
  v4f vv = (pos < clen) ? __builtin_nontemporal_load(&vin[idx]) : z;
  __builtin_nontemporal_store(kv, &kout[idx]);
  __builtin_nontemporal_store(vv, &vout[idx]);
}

// ---------------- embedding + positional ----------------
__global__ void embed_kernel(const int* __restrict__ tok, const int* __restrict__ pos,
                             const float* __restrict__ emb, const float* __restrict__ pe,
                             const float* __restrict__ xscale, const float* __restrict__ alpha,
                             float* __restrict__ x) {
  int i = blockIdx.x * 256 + threadIdx.x;               // grid 2 -> 512
  int t = tok[0], p = pos[0];
  x[i] = emb[(size_t)t * HID + i] * xscale[0] + alpha[0] * pe[(size_t)p * HID + i];
}

// ---------------- QKV GEMV: 512 -> 1536, routes k/v into the output cache row clen ----
__global__ void qkv_kernel(const float* __restrict__ x, const float* __restrict__ W,
                           const float* __restrict__ bias, float* __restrict__ q,
                           float* __restrict__ kout_l, float* __restrict__ vout_l,
                           const int* __restrict__ clen_p) {
  __shared__ float xs[HID];
  int tid = threadIdx.x;
  int col = blockIdx.x * 256 + tid;                      // grid 6 -> 1536
  for (int i = tid; i < HID; i += 256) xs[i] = x[i];
  __syncthreads();
  float a0 = 0.f, a1 = 0.f, a2 = 0.f, a3 = 0.f;
  const float* wp = W + col;
  for (int i = 0; i < HID; i += 4) {
    a0 += xs[i + 0] * __builtin_nontemporal_load(&wp[(i + 0) * 1536]);
    a1 += xs[i + 1] * __builtin_nontemporal_load(&wp[(i + 1) * 1536]);
    a2 += xs[i + 2] * __builtin_nontemporal_load(&wp[(i + 2) * 1536]);
    a3 += xs[i + 3] * __builtin_nontemporal_load(&wp[(i + 3) * 1536]);
  }
  float acc = (a0 + a1) + (a2 + a3) + bias[col];
  int clen = clen_p[0];
  if (col < HID)           q[col] = acc;
  else if (col < 2 * HID)  kout_l[(size_t)clen * HID + (col - HID)] = acc;
  else                     vout_l[(size_t)clen * HID + (col - 2 * HID)] = acc;
}

// ---------------- attention: one block per head; WMMA f32 16x16x4 for both QK^T and AV ----
__global__ void attn_kernel(const float* __restrict__ q, const float* __restrict__ kc,
                            const float* __restrict__ vc, float* __restrict__ o_out,
                            const int* __restrict__ clen_p) {
  const int h = blockIdx.x;
  const int tid = threadIdx.x;
  const int wave = tid >> 5;
  const int lane = tid & 31;
  const int next_len = clen_p[0] + 1;
  const int nchunks = (next_len + 15) >> 4;
  const float scale = 0.17677669529663687f;              // 1/sqrt(32)
  const int off = (lane >> 4) << 1;                      // 0 (lanes 0-15) or 2 (lanes 16-31)
  const int n = lane & 15;
  const float* qh = q + h * HDIM;

  // A fragments for QK: all 16 rows replicated with q -> every D row holds the scores.
  // 32-bit A 16x4 layout: lane m<16 -> {A[m][0],A[m][1]}, lane m+16 -> {A[m][2],A[m][3]}
  v2f a[8];
#pragma unroll
  for (int j = 0; j < 8; j++) {
    a[j].x = qh[4 * j + off]     * scale;
    a[j].y = qh[4 * j + off + 1] * scale;
  }

  float m = -INFINITY, s = 0.f;
  v8f cav0 = {0.f, 0.f, 0.f, 0.f, 0.f, 0.f, 0.f, 0.f};  // AV accum, dims 0..15
  v8f cav1 = {0.f, 0.f, 0.f, 0.f, 0.f, 0.f, 0.f, 0.f};  // AV accum, dims 16..31

  for (int cb = wave; cb < nchunks; cb += 8) {
    const int kbase = cb << 4;
    const int key = kbase + n;
    // ---- QK^T scores via 8 chained WMMAs (K = 32) ----
    // B 4x16 layout: lane n<16 -> {B[0][n],B[1][n]}, lane n+16 -> {B[2][n],B[3][n]}
    const float* kr = kc + (size_t)key * HID + h * HDIM + off;
    v8f c = {0.f, 0.f, 0.f, 0.f, 0.f, 0.f, 0.f, 0.f};
#pragma unroll
    for (int j = 0; j < 8; j++) {
      v2f b;
      b.x = kr[4 * j];
      b.y = kr[4 * j + 1];
      c = __builtin_amdgcn_wmma_f32_16x16x4_f32(false, a[j], false, b, (short)0, c,
                                                false, false);
    }
    // c[0] holds score[key kbase + (lane&15)] duplicated across both halves
    float sc = (key < next_len) ? c[0] : -1e30f;
    float cmax = half16Max(sc);
    float newm = fmaxf(m, cmax);
    float corr = __expf(m - newm);
    float p = __expf(sc - newm);            // duplicated across halves
    float psum = half16Sum(p);
    s = s * corr + psum;
#pragma unroll
    for (int e = 0; e < 8; e++) { cav0[e] *= corr; cav1[e] *= corr; }

    // ---- AV via 8 WMMAs: P(1x16, replicated rows) x V(16x32) ----
    const float* vr = vc + (size_t)kbase * HID + h * HDIM;
#pragma unroll
    for (int g = 0; g < 4; g++) {
      v2f pa;
      pa.x = __shfl(p, 4 * g + off, 32);
      pa.y = __shfl(p, 4 * g + off + 1, 32);
      const float* v0 = vr + (size_t)(4 * g + off) * HID + n;   // rows 4g+off, 4g+off+1
      v2f b0; b0.x = v0[0];  b0.y = v0[HID];                    // dims n (0..15)
      cav0 = __builtin_amdgcn_wmma_f32_16x16x4_f32(false, pa, false, b0, (short)0, cav0,
                                                   false, false);
      v2f b1; b1.x = v0[16]; b1.y = v0[HID + 16];               // dims 16+n
      cav1 = __builtin_amdgcn_wmma_f32_16x16x4_f32(false, pa, false, b1, (short)0, cav1,
                                                   false, false);
    }
    m = newm;
  }

  // row 0 of D: lanes 0-15 give dims N=lane; cav1 lanes 16-31 give dims 16+(lane-16)=lane
  float o = (lane < 16) ? cav0[0] : cav1[0];

  __shared__ float m_s[8], s_s[8], o_s[8 * 32];
  if (lane == 0) { m_s[wave] = m; s_s[wave] = s; }
  o_s[wave * 32 + lane] = o;
  __syncthreads();
  if (wave == 0) {
    float M = -INFINITY;
#pragma unroll
    for (int w = 0; w < 8; w++) M = fmaxf(M, m_s[w]);
    float T = 0.f, od = 0.f;
#pragma unroll
    for (int w = 0; w < 8; w++) {
      float f = __expf(m_s[w] - M);
      T  += s_s[w] * f;
      od += o_s[w * 32 + lane] * f;
    }
    o_out[h * HDIM + lane] = od / T;
  }
}

// ---------------- generic coalesced GEMV: out[col] = dot(x, W[:,col]) + b, optional ReLU ----
template <bool RELU>
__global__ void gemv_kernel(const float* __restrict__ x, const float* __restrict__ W,
                            const float* __restrict__ bias, float* __restrict__ out,
                            int IC, int OC) {
  __shared__ float xs[FF];
  int tid = threadIdx.x;
  int col = blockIdx.x * 256 + tid;
  for (int i = tid; i < IC; i += 256) xs[i] = x[i];
  __syncthreads();
  float a0 = 0.f, a1 = 0.f, a2 = 0.f, a3 = 0.f;
  const float* wp = W + col;
  for (int i = 0; i < IC; i += 4) {
    a0 += xs[i + 0] * __builtin_nontemporal_load(&wp[(i + 0) * OC]);
    a1 += xs[i + 1] * __builtin_nontemporal_load(&wp[(i + 1) * OC]);
    a2 += xs[i + 2] * __builtin_nontemporal_load(&wp[(i + 2) * OC]);
    a3 += xs[i + 3] * __builtin_nontemporal_load(&wp[(i + 3) * OC]);
  }
  float acc = (a0 + a1) + (a2 + a3) + bias[col];
  if (RELU) acc = fmaxf(acc, 0.f);
  out[col] = acc;
}

// ---------------- layer norm with residual: x = LN(x + add) * w + b (single block, 256) ----
__global__ void ln_kernel(float* __restrict__ x, const float* __restrict__ add,
                          const float* __restrict__ w, const float* __restrict__ b) {
  int tid = threadIdx.x;
  float v0 = x[tid] + add[tid];
  float v1 = x[tid + 256] + add[tid + 256];
  float mu = blockSum256(v0 + v1) * (1.f / 512.f);
  float d0 = v0 - mu, d1 = v1 - mu;
  float var = blockSum256(d0 * d0 + d1 * d1) * (1.f / 512.f);
  float rs = rsqrtf(var + 1e-5f);
  x[tid]       = d0 * rs * w[tid] + b[tid];
  x[tid + 256] = d1 * rs * w[tid + 256] + b[tid + 256];
}

// ---------------- logits: one wave per vocab row ----------------
__global__ void logits_kernel(const float* __restrict__ x, const float* __restrict__ pw,
                              const float* __restrict__ pb, float* __restrict__ logits) {
  int wave = threadIdx.x >> 5, lane = threadIdx.x & 31;
  int row = blockIdx.x * 8 + wave;
  if (row >= VOCAB) return;
  const float* wr = pw + (size_t)row * HID;
  float acc = 0.f;
#pragma unroll 4
  for (int d = lane; d < HID; d += 32) acc += x[d] * __builtin_nontemporal_load(&wr[d]);
  acc = waveSum(acc);
  if (lane == 0) logits[row] = acc + pb[row];
}

// ---------------- argmax + scalar outputs ----------------
__global__ void finalize_kernel(const float* __restrict__ logits, const int* __restrict__ clen_p,
                                const int* __restrict__ pos_p, int* __restrict__ iout) {
  __shared__ float bv[256];
  __shared__ int   bi[256];
  int tid = threadIdx.x;
  float best = -INFINITY;
  int bidx = 0x7fffffff;
  for (int i = tid; i < VOCAB; i += 256) {
    float v = logits[i];
    if (v > best || (v == best && i < bidx)) { best = v; bidx = i; }
  }
  bv[tid] = best; bi[tid] = bidx;
  __syncthreads();
  for (int sft = 128; sft > 0; sft >>= 1) {
    if (tid < sft) {
      if (bv[tid + sft] > bv[tid] || (bv[tid + sft] == bv[tid] && bi[tid + sft] < bi[tid])) {
        bv[tid] = bv[tid + sft]; bi[tid] = bi[tid + sft];
      }
    }
    __syncthreads();
  }
  if (tid == 0) {
    iout[0] = bi[0];                 // sampled
    iout[1] = (bi[0] == 1024);       // eos
    iout[2] = clen_p[0] + 1;         // next_len
    iout[3] = pos_p[0] + 1;          // next_pos
  }
}

// ---------------- host orchestration ----------------
extern "C" void kernel_launch(void* const* d_in, const int* in_sizes, int n_in,
                              void* d_out, int out_size, void* d_ws, size_t ws_size,
                              hipStream_t stream) {
  (void)in_sizes; (void)n_in; (void)out_size; (void)ws_size;
  const int*   last_token = (const int*)d_in[0];
  const int*   pos_index  = (const int*)d_in[1];
  const int*   cache_len  = (const int*)d_in[2];
  const float* k_cache    = (const float*)d_in[3];
  const float* v_cache    = (const float*)d_in[4];
  const float* emb_table  = (const float*)d_in[5];
  const float* pe_table   = (const float*)d_in[6];
  const float* x_scale    = (const float*)d_in[7];
  const float* alpha      = (const float*)d_in[8];
  const float* qkv_w_t    = (const float*)d_in[9];
  const float* qkv_b      = (const float*)d_in[10];
  const float* out_w_t    = (const float*)d_in[11];
  const float* out_b      = (const float*)d_in[12];
  const float* norm_w1    = (const float*)d_in[13];
  const float* norm_b1    = (const float*)d_in[14];
  const float* norm_w2    = (const float*)d_in[15];
  const float* norm_b2    = (const float*)d_in[16];
  const float* w1_t       = (const float*)d_in[17];
  const float* b1         = (const float*)d_in[18];
  const float* w2_t       = (const float*)d_in[19];
  const float* b2         = (const float*)d_in[20];
  const float* pred_w     = (const float*)d_in[21];
  const float* pred_b     = (const float*)d_in[22];

  // outputs: logits(1025) | sampled(1) | eos(1) | next_len(1) | next_pos(1) | k_out | v_out
  float* out_f  = (float*)d_out;
  float* logits = out_f;
  int*   iout   = (int*)d_out + VOCAB;
  float* k_out  = out_f + VOCAB + 4;
  float* v_out  = k_out + KV_ELEMS;

  // workspace layout (floats)
  float* ws     = (float*)d_ws;
  float* x      = ws;            // 512
  float* q      = ws + 512;      // 512
  float* attn_o = ws + 1024;     // 512
  float* o_proj = ws + 1536;     // 512
  float* h      = ws + 2048;     // 2048
  float* h2     = ws + 4096;     // 512

  // 1) bulk KV copy/zero (16B granularity, k and v together)
  {
    int blocks = (int)((KV_ELEMS / 4) / 256);            // 24576
    kv_copy_kernel<<<blocks, 256, 0, stream>>>(
        (const v4f*)k_cache, (const v4f*)v_cache,
        (v4f*)k_out, (v4f*)v_out, cache_len);
  }

  // 2) embedding + positional
  embed_kernel<<<2, 256, 0, stream>>>(last_token, pos_index, emb_table, pe_table,
                                      x_scale, alpha, x);

  // 3) layers
  for (int l = 0; l < N_LAYERS; ++l) {
    const float* qw  = qkv_w_t + (size_t)l * HID * (3 * HID);
    const float* qb  = qkv_b   + (size_t)l * (3 * HID);
    const float* ow  = out_w_t + (size_t)l * HID * HID;
    const float* ob  = out_b   + (size_t)l * HID;
    const float* W1  = w1_t    + (size_t)l * HID * FF;
    const float* B1  = b1      + (size_t)l * FF;
    const float* W2  = w2_t    + (size_t)l * FF * HID;
    const float* B2  = b2      + (size_t)l * HID;
    float* kl = k_out + (size_t)l * CAP * HID;
    float* vl = v_out + (size_t)l * CAP * HID;

    qkv_kernel<<<6, 256, 0, stream>>>(x, qw, qb, q, kl, vl, cache_len);
    attn_kernel<<<HEADS, 256, 0, stream>>>(q, kl, vl, attn_o, cache_len);
    gemv_kernel<false><<<HID / 256, 256, 0, stream>>>(attn_o, ow, ob, o_proj, HID, HID);
    ln_kernel<<<1, 256, 0, stream>>>(x, o_proj, norm_w1 + (size_t)l * HID,
                                     norm_b1 + (size_t)l * HID);
    gemv_kernel<true><<<FF / 256, 256, 0, stream>>>(x, W1, B1, h, HID, FF);
    gemv_kernel<false><<<HID / 256, 256, 0, stream>>>(h, W2, B2, h2, FF, HID);
    ln_kernel<<<1, 256, 0, stream>>>(x, h2, norm_w2 + (size_t)l * HID,
                                     norm_b2 + (size_t)l * HID);
  }

  // 4) prediction head + argmax/scalars
  logits_kernel<<<(VOCAB + 7) / 8, 256, 0, stream>>>(x, pred_w, pred_b, logits);
  finalize_kernel<<<1, 256, 0, stream>>>(logits, cache_len, pos_index, iout);
}